// OptNetLatent_Q_54606214201657
// MI455X (gfx1250) — compile-verified
//
#include <hip/hip_runtime.h>
#include <math.h>

#define B_    512
#define NIN   256
#define NZ    128
#define NOUT  256
#define NEQ   32
#define NINEQ 128
#define NITER 18
#define SIGMA 0.1f

typedef __attribute__((ext_vector_type(2))) float v2f;
typedef __attribute__((ext_vector_type(8))) float v8f;

// D = A(16x4 f32) * B(4x16 f32) + C(16x16 f32), exact-precision tensor op.
__device__ __forceinline__ v8f wmma4(v2f a, v2f b, v8f c) {
  return __builtin_amdgcn_wmma_f32_16x16x4_f32(
      /*neg_a=*/false, a, /*neg_b=*/false, b,
      /*c_mod=*/(short)0, c, /*reuse_a=*/false, /*reuse_b=*/false);
}

// ---------------------------------------------------------------- transpose G
__global__ __launch_bounds__(256) void k_transpose(const float* __restrict__ G,
                                                   float* __restrict__ Gt) {
  int idx = blockIdx.x * 256 + threadIdx.x;   // NZ*NZ = 16384
  int m = idx >> 7, i = idx & 127;
  Gt[i * NZ + m] = G[m * NZ + i];
}

// ------------------------------------------------- p = relu(x @ W_in^T + b_in)
__global__ __launch_bounds__(256) void k_in(const float* __restrict__ x,
                                            const float* __restrict__ Win,
                                            const float* __restrict__ bin,
                                            float* __restrict__ p) {
  int lane = threadIdx.x & 31, wv = threadIdx.x >> 5;
  int t = blockIdx.x * 8 + wv;          // 256 tiles: 32 row-tiles x 8 col-tiles
  int ti = t >> 3, tj = t & 7;
  int mi = lane & 15;
  int koff = (lane >> 4) << 1;          // A/B 16x4 f32 layout: K halves split on lane halves
  const float* xrow = x + (size_t)(ti * 16 + mi) * NIN;
  const float* wrow = Win + (size_t)(tj * 16 + mi) * NIN;
  v8f acc = {};
  for (int k0 = 0; k0 < NIN; k0 += 4) {
    v2f a = *(const v2f*)(xrow + k0 + koff);
    v2f b = *(const v2f*)(wrow + k0 + koff);
    acc = wmma4(a, b, acc);
  }
  int rbase = ti * 16 + ((lane >> 4) << 3);
  int col = tj * 16 + mi;
  float bias = bin[col];
#pragma unroll
  for (int r = 0; r < 8; ++r) {
    float v = acc[r] + bias;
    p[(size_t)(rbase + r) * NZ + col] = v > 0.f ? v : 0.f;
  }
}

// ------------------------------------------- out = sigmoid(z @ W_out^T + b_out)
__global__ __launch_bounds__(256) void k_out(const float* __restrict__ z,
                                             const float* __restrict__ Wout,
                                             const float* __restrict__ bout,
                                             float* __restrict__ out) {
  int lane = threadIdx.x & 31, wv = threadIdx.x >> 5;
  int t = blockIdx.x * 8 + wv;          // 512 tiles: 32 x 16
  int ti = t >> 4, tj = t & 15;
  int mi = lane & 15;
  int koff = (lane >> 4) << 1;
  const float* zr = z + (size_t)(ti * 16 + mi) * NZ;
  const float* wr = Wout + (size_t)(tj * 16 + mi) * NZ;
  v8f acc = {};
  for (int k0 = 0; k0 < NZ; k0 += 4) {
    v2f a = *(const v2f*)(zr + k0 + koff);
    v2f b = *(const v2f*)(wr + k0 + koff);
    acc = wmma4(a, b, acc);
  }
  int rbase = ti * 16 + ((lane >> 4) << 3);
  int col = tj * 16 + mi;
  float bias = bout[col];
#pragma unroll
  for (int r = 0; r < 8; ++r) {
    float v = acc[r] + bias;
    out[(size_t)(rbase + r) * NOUT + col] = 1.f / (1.f + expf(-v));
  }
}

// ------------------------------------------------------------- QP IPM solver
// One workgroup per batch element. Schur-complement KKT solve:
//   H dz + A^T dy = rhs_z ;  A dz = -ry
//   H SPD -> chol(H); V = H^{-1}A^T, u = H^{-1}rhs_z; S = A V; dy = S^{-1}(A u + ry);
//   dz = u - V dy.
__global__ __launch_bounds__(256) void k_qp(const float* __restrict__ pbuf,
                                            const float* __restrict__ h1,
                                            const float* __restrict__ Qg,
                                            const float* __restrict__ Gg,
                                            const float* __restrict__ Gtg,
                                            const float* __restrict__ Ag,
                                            const float* __restrict__ bg,
                                            float* __restrict__ zbuf) {
  __shared__ float sH[NZ * NZ];        // H, then its Cholesky factor L
  __shared__ float sV[NZ * NEQ];       // H^{-1} A^T (33-RHS triangular solves)
  __shared__ float sS[NEQ * NEQ];      // Schur complement
  __shared__ float sz[NZ], sp[NZ], srz[NZ], su[NZ], sdz[NZ];
  __shared__ float slam[NINEQ], ss_[NINEQ], sd[NINEQ], sh[NINEQ];
  __shared__ float srp[NINEQ], sw[NINEQ], sdlam[NINEQ], sds[NINEQ];
  __shared__ float sy[NEQ], sry[NEQ], sdy[NEQ];
  __shared__ float sred[256];

  const int tid = threadIdx.x;
  const int bid = blockIdx.x;

  // Warm L2/WGP$ with the shared matrices (gfx1250 global_prefetch_b8).
  if (tid < 128) {
    __builtin_prefetch(Qg + tid * NZ, 0, 0);
    __builtin_prefetch(Gg + tid * NZ, 0, 0);
    __builtin_prefetch(Gtg + tid * NZ, 0, 0);
  }

  // init state: z=0, y=0, s=h, lam=1
  if (tid < NZ) { sz[tid] = 0.f; sp[tid] = pbuf[(size_t)bid * NZ + tid]; }
  if (tid < NINEQ) {
    float hv = h1[(size_t)bid * NINEQ + tid];
    sh[tid] = hv; ss_[tid] = hv; slam[tid] = 1.f;
  }
  if (tid < NEQ) sy[tid] = 0.f;
  __syncthreads();

  for (int it = 0; it < NITER; ++it) {
    // ---- Phase A: residuals rz (tid<128) ; d, rp (tid>=128) ; mu partials
    if (tid < NZ) {
      const float* qrow = Qg + (size_t)tid * NZ;
      const float* gtr = Gtg + (size_t)tid * NZ;
      float acc = sp[tid];
      for (int j = 0; j < NZ; ++j) acc += qrow[j] * sz[j];
      for (int q = 0; q < NEQ; ++q) acc += sy[q] * Ag[q * NZ + tid];
      for (int m = 0; m < NINEQ; ++m) acc += slam[m] * gtr[m];
      srz[tid] = acc;
    } else {
      int i = tid - NZ;
      sd[i] = slam[i] / ss_[i];
      const float* grow = Gg + (size_t)i * NZ;
      float acc = ss_[i] - sh[i];
      for (int j = 0; j < NZ; ++j) acc += grow[j] * sz[j];
      srp[i] = acc;
    }
    sred[tid] = (tid < NINEQ) ? slam[tid] * ss_[tid] : 0.f;
    __syncthreads();

    // ---- Phase B: ry ; mu reduction
    if (tid < NEQ) {
      const float* arow = Ag + (size_t)tid * NZ;
      float acc = -bg[tid];
      for (int j = 0; j < NZ; ++j) acc += arow[j] * sz[j];
      sry[tid] = acc;
    }
    for (int off = 128; off > 0; off >>= 1) {
      if (tid < off) sred[tid] += sred[tid + off];
      __syncthreads();
    }
    float mu = sred[0] * (1.0f / NINEQ);

    // ---- Phase C: w = (lam*rp - (lam*s - sigma*mu)) / s
    if (tid < NINEQ) {
      float rc = slam[tid] * ss_[tid] - SIGMA * mu;
      sw[tid] = (slam[tid] * srp[tid] - rc) / ss_[tid];
    }
    __syncthreads();

    // ---- Phase D: u <- rhs_z = -(rz + G^T w) ; init V = A^T
    if (tid < NZ) {
      const float* gtr = Gtg + (size_t)tid * NZ;
      float acc = srz[tid];
      for (int m = 0; m < NINEQ; ++m) acc += sw[m] * gtr[m];
      su[tid] = -acc;
    }
    for (int idx = tid; idx < NZ * NEQ; idx += 256) {
      int i = idx >> 5, r = idx & 31;           // idx = i*NEQ + r
      sV[idx] = Ag[r * NZ + i];
    }

    // ---- Phase E (WMMA): H = Q + G^T diag(d) G ; 64 16x16 tiles, K=128
    {
      int lane = tid & 31, wv = tid >> 5;
      int mi = lane & 15;
      int koff = (lane >> 4) << 1;
      for (int t = wv; t < 64; t += 8) {
        int ti = t >> 3, tj = t & 7;
        int i = ti * 16 + mi, j = tj * 16 + mi;
        const float* gtr = Gtg + (size_t)i * NZ;
        v8f acc = {};
        for (int k0 = 0; k0 < NZ; k0 += 4) {
          int m = k0 + koff;
          v2f ga = *(const v2f*)(gtr + m);       // A[i,m] = G[m,i]*d_m
          v2f a; a.x = ga.x * sd[m]; a.y = ga.y * sd[m + 1];
          v2f bb;                                // B[m,j] = G[m,j]
          bb.x = Gg[(size_t)m * NZ + j];
          bb.y = Gg[(size_t)(m + 1) * NZ + j];
          acc = wmma4(a, bb, acc);
        }
        int rbase = ti * 16 + ((lane >> 4) << 3);
#pragma unroll
        for (int r = 0; r < 8; ++r) {
          int row = rbase + r;
          sH[row * NZ + j] = acc[r] + Qg[(size_t)row * NZ + j];
        }
      }
    }
    __syncthreads();

    // ---- Cholesky of H (lower triangle, in place)
    for (int k = 0; k < NZ; ++k) {
      if (tid == 0) sH[k * NZ + k] = sqrtf(sH[k * NZ + k]);
      __syncthreads();
      float dk = sH[k * NZ + k];
      for (int i = k + 1 + tid; i < NZ; i += 256) sH[i * NZ + k] /= dk;
      __syncthreads();
      int rem = NZ - 1 - k;
      for (int idx = tid; idx < rem * rem; idx += 256) {
        int ii = k + 1 + idx / rem, jj = k + 1 + idx % rem;
        if (jj <= ii) sH[ii * NZ + jj] -= sH[ii * NZ + k] * sH[jj * NZ + k];
      }
      __syncthreads();
    }

    // ---- Forward solve L * Y = [A^T | rhs_z]  (33 RHS: V cols + u)
    for (int k = 0; k < NZ; ++k) {
      float lkk = sH[k * NZ + k];
      if (tid < NEQ) sV[k * NEQ + tid] /= lkk;
      else if (tid == NEQ) su[k] /= lkk;
      __syncthreads();
      int rows = NZ - 1 - k;
      for (int idx = tid; idx < rows * (NEQ + 1); idx += 256) {
        int i = k + 1 + idx / (NEQ + 1);
        int r = idx % (NEQ + 1);
        if (r < NEQ) sV[i * NEQ + r] -= sH[i * NZ + k] * sV[k * NEQ + r];
        else su[i] -= sH[i * NZ + k] * su[k];
      }
      __syncthreads();
    }
    // ---- Backward solve L^T * X = Y
    for (int k = NZ - 1; k >= 0; --k) {
      float lkk = sH[k * NZ + k];
      if (tid < NEQ) sV[k * NEQ + tid] /= lkk;
      else if (tid == NEQ) su[k] /= lkk;
      __syncthreads();
      for (int idx = tid; idx < k * (NEQ + 1); idx += 256) {
        int i = idx / (NEQ + 1);
        int r = idx % (NEQ + 1);
        if (r < NEQ) sV[i * NEQ + r] -= sH[k * NZ + i] * sV[k * NEQ + r];
        else su[i] -= sH[k * NZ + i] * su[k];
      }
      __syncthreads();
    }

    // ---- Schur: S = A V ; rhs_y = A u + ry
    for (int idx = tid; idx < NEQ * NEQ; idx += 256) {
      int pp = idx / NEQ, q = idx % NEQ;
      const float* arow = Ag + (size_t)pp * NZ;
      float acc = 0.f;
      for (int i = 0; i < NZ; ++i) acc += arow[i] * sV[i * NEQ + q];
      sS[idx] = acc;
    }
    if (tid < NEQ) {
      const float* arow = Ag + (size_t)tid * NZ;
      float acc = sry[tid];
      for (int i = 0; i < NZ; ++i) acc += arow[i] * su[i];
      sdy[tid] = acc;
    }
    __syncthreads();

    // ---- Cholesky of S (32x32)
    for (int k = 0; k < NEQ; ++k) {
      if (tid == 0) sS[k * NEQ + k] = sqrtf(sS[k * NEQ + k]);
      __syncthreads();
      float dk = sS[k * NEQ + k];
      for (int i = k + 1 + tid; i < NEQ; i += 256) sS[i * NEQ + k] /= dk;
      __syncthreads();
      int rem = NEQ - 1 - k;
      for (int idx = tid; idx < rem * rem; idx += 256) {
        int ii = k + 1 + idx / rem, jj = k + 1 + idx % rem;
        if (jj <= ii) sS[ii * NEQ + jj] -= sS[ii * NEQ + k] * sS[jj * NEQ + k];
      }
      __syncthreads();
    }
    // ---- dy = S^{-1} rhs_y (small: serial on one lane)
    if (tid == 0) {
      for (int k = 0; k < NEQ; ++k) {
        float acc = sdy[k];
        for (int j = 0; j < k; ++j) acc -= sS[k * NEQ + j] * sdy[j];
        sdy[k] = acc / sS[k * NEQ + k];
      }
      for (int k = NEQ - 1; k >= 0; --k) {
        float acc = sdy[k];
        for (int j = k + 1; j < NEQ; ++j) acc -= sS[j * NEQ + k] * sdy[j];
        sdy[k] = acc / sS[k * NEQ + k];
      }
    }
    __syncthreads();

    // ---- dz = u - V dy
    if (tid < NZ) {
      float acc = su[tid];
      for (int q = 0; q < NEQ; ++q) acc -= sV[tid * NEQ + q] * sdy[q];
      sdz[tid] = acc;
    }
    __syncthreads();

    // ---- Gdz, dlam, ds
    if (tid < NINEQ) {
      const float* grow = Gg + (size_t)tid * NZ;
      float acc = 0.f;
      for (int j = 0; j < NZ; ++j) acc += grow[j] * sdz[j];
      sdlam[tid] = sd[tid] * acc + sw[tid];
      sds[tid] = -srp[tid] - acc;
    }
    __syncthreads();

    // ---- alpha = min(1, 0.99 * min(ftb(s,ds), ftb(lam,dlam)))
    float cand = 1e10f;
    if (tid < NINEQ) {
      float c1 = (sds[tid] < 0.f) ? (-ss_[tid] / sds[tid]) : 1e10f;
      float c2 = (sdlam[tid] < 0.f) ? (-slam[tid] / sdlam[tid]) : 1e10f;
      cand = fminf(c1, c2);
    }
    sred[tid] = cand;
    __syncthreads();
    for (int off = 128; off > 0; off >>= 1) {
      if (tid < off) sred[tid] = fminf(sred[tid], sred[tid + off]);
      __syncthreads();
    }
    float alpha = fminf(1.f, 0.99f * sred[0]);

    // ---- state update
    if (tid < NZ) sz[tid] += alpha * sdz[tid];
    if (tid < NEQ) sy[tid] += alpha * sdy[tid];
    if (tid < NINEQ) {
      slam[tid] += alpha * sdlam[tid];
      ss_[tid] += alpha * sds[tid];
    }
    __syncthreads();
  }

  if (tid < NZ) zbuf[(size_t)bid * NZ + tid] = sz[tid];
}

// ---------------------------------------------------------------------------
extern "C" void kernel_launch(void* const* d_in, const int* in_sizes, int n_in,
                              void* d_out, int out_size, void* d_ws, size_t ws_size,
                              hipStream_t stream) {
  const float* x    = (const float*)d_in[0];   // [512,256]
  const float* h1   = (const float*)d_in[1];   // [512,128]
  const float* Q1   = (const float*)d_in[2];   // [128,128]
  const float* G1   = (const float*)d_in[3];   // [128,128]
  const float* A1   = (const float*)d_in[4];   // [32,128]
  const float* b1   = (const float*)d_in[5];   // [32]
  const float* Win  = (const float*)d_in[6];   // [128,256]
  const float* bin  = (const float*)d_in[7];   // [128]
  const float* Wout = (const float*)d_in[8];   // [256,128]
  const float* bout = (const float*)d_in[9];   // [256]
  float* out = (float*)d_out;                  // [512,256]

  float* p_buf = (float*)d_ws;                 // B_*NZ
  float* z_buf = p_buf + (size_t)B_ * NZ;      // B_*NZ
  float* Gt    = z_buf + (size_t)B_ * NZ;      // NZ*NZ

  k_transpose<<<(NZ * NZ) / 256, 256, 0, stream>>>(G1, Gt);
  k_in<<<(B_ / 16) * (NZ / 16) / 8, 256, 0, stream>>>(x, Win, bin, p_buf);
  k_qp<<<B_, 256, 0, stream>>>(p_buf, h1, Q1, G1, Gt, A1, b1, z_buf);
  k_out<<<(B_ / 16) * (NOUT / 16) / 8, 256, 0, stream>>>(z_buf, Wout, bout, out);
}